// RpnToProposal_73787538145733
// MI455X (gfx1250) — compile-verified
//
#include <hip/hip_runtime.h>
#include <hip/hip_bf16.h>

// RPN -> proposals for MI455X (gfx1250, wave32).
// B=2, N=131072, 2 classes, OUT=2000. Greedy NMS == greedy-over-sorted-scores,
// so: decode+score (stream), top-K=4096 select+sort, WMMA-assisted KxK pairwise
// suppression bitmatrix (async-to-LDS staged tiles, 8 WMMAs per wave strip),
// then a cheap serial bitmask scan.

#define BB 2
#define NN 131072          // 2^17
#define KK 4096            // candidate pool (power of two for bitonic)
#define KW (KK/16)         // 256 uint16 words per suppression row
#define OUTN 2000
#define SCORE_THR 0.05f
#define IOU_THR 0.7f
#define NEGV -1000000000.0f

typedef __attribute__((ext_vector_type(2))) float v2f;
typedef __attribute__((ext_vector_type(8))) float v8f;

__device__ __forceinline__ int bucketOf(float s) {
    int v = (int)(s * 256.0f);
    return v < 0 ? 0 : (v > 255 ? 255 : v);
}

// ---------------------------------------------------------------- init
__global__ void rpn_init_kernel(float* out, unsigned int* hist, unsigned int* cnt) {
    int g = blockIdx.x * 256 + threadIdx.x;
    if (g < BB * OUTN * 10) out[g] = 0.0f;     // 40000 output floats -> zero (invalid slots)
    if (g < BB * 256) hist[g] = 0u;
    if (g < BB) cnt[g] = 0u;
}

// ---------------------------------------------------------------- decode + score + histogram
__global__ __launch_bounds__(256) void rpn_decode_kernel(
    const float4* __restrict__ deltas, const float* __restrict__ logits,
    const float4* __restrict__ anchors, float4* __restrict__ boxes,
    float* __restrict__ live, unsigned int* __restrict__ hist) {
    int g = blockIdx.x * 256 + threadIdx.x;
    if (g >= BB * NN) return;
    if (g + 4096 < BB * NN) __builtin_prefetch(&deltas[g + 4096], 0, 1); // global_prefetch_b8
    float4 dd = deltas[g];
    float4 aa = anchors[g];
    float l0 = logits[(size_t)g * 2 + 0];
    float l1 = logits[(size_t)g * 2 + 1];
    // softmax over 2 classes; fg prob = sigmoid(l1 - l0)
    float score = 1.0f / (1.0f + __expf(l0 - l1));
    float h  = aa.z - aa.x, w  = aa.w - aa.y;
    float cy = (aa.z + aa.x) * 0.5f, cx = (aa.w + aa.y) * 0.5f;
    cy += dd.x * 0.1f * h;
    cx += dd.y * 0.1f * w;
    h *= __expf(dd.z * 0.2f);
    w *= __expf(dd.w * 0.2f);
    boxes[g] = make_float4(cy - h * 0.5f, cx - w * 0.5f, cy + h * 0.5f, cx + w * 0.5f);
    float lv = score > SCORE_THR ? score : NEGV;
    live[g] = lv;
    if (lv > 0.0f) {
        int b = g >> 17;
        atomicAdd(&hist[b * 256 + bucketOf(lv)], 1u);
    }
}

// ---------------------------------------------------------------- pick score cutoff bucket
__global__ void rpn_cutoff_kernel(const unsigned int* __restrict__ hist,
                                  unsigned int* __restrict__ cutoff) {
    int b = threadIdx.x;
    if (b >= BB) return;
    unsigned cum = 0, cut = 0;
    for (int k = 255; k >= 0; --k) {
        cum += hist[b * 256 + k];
        if (cum >= KK) { cut = (unsigned)k; break; }
    }
    cutoff[b] = cut;
}

// ---------------------------------------------------------------- compact candidates
__global__ __launch_bounds__(256) void rpn_compact_kernel(
    const float* __restrict__ live, const unsigned int* __restrict__ cutoff,
    unsigned int* __restrict__ cnt, unsigned long long* __restrict__ ckey) {
    int g = blockIdx.x * 256 + threadIdx.x;
    if (g >= BB * NN) return;
    int b = g >> 17;
    int idx = g & (NN - 1);
    float lv = live[g];
    if (lv > 0.0f && bucketOf(lv) >= (int)cutoff[b]) {
        unsigned pos = atomicAdd(&cnt[b], 1u);
        if (pos < KK) {
            // key: [score bits | ~idx] -> descending sort gives score desc, idx asc on ties
            unsigned long long key =
                ((unsigned long long)__float_as_uint(lv) << 32) | (unsigned)(~idx);
            ckey[(size_t)b * KK + pos] = key;
        }
    }
}

// ---------------------------------------------------------------- in-LDS bitonic sort (desc)
__global__ __launch_bounds__(1024) void rpn_sort_kernel(
    const unsigned long long* __restrict__ ckey, const unsigned int* __restrict__ cnt,
    const float4* __restrict__ boxes, float* __restrict__ cscore,
    unsigned int* __restrict__ corig, float4* __restrict__ cbox) {
    int b = blockIdx.x;
    int tid = threadIdx.x;
    __shared__ unsigned long long s[KK];   // 32 KB LDS
    unsigned c = cnt[b];
    int count = c < KK ? (int)c : KK;
    for (int i = tid; i < KK; i += 1024) s[i] = (i < count) ? ckey[(size_t)b * KK + i] : 0ull;
    __syncthreads();
    for (int k = 2; k <= KK; k <<= 1) {
        for (int j = k >> 1; j > 0; j >>= 1) {
            for (int i = tid; i < KK; i += 1024) {
                int ixj = i ^ j;
                if (ixj > i) {
                    unsigned long long A = s[i], Bv = s[ixj];
                    bool desc = ((i & k) == 0);           // overall descending
                    if (desc ? (A < Bv) : (A > Bv)) { s[i] = Bv; s[ixj] = A; }
                }
            }
            __syncthreads();
        }
    }
    for (int r = tid; r < KK; r += 1024) {
        unsigned long long key = s[r];
        unsigned hi = (unsigned)(key >> 32);
        size_t o = (size_t)b * KK + r;
        if (hi == 0u) {  // padding slot
            cscore[o] = NEGV;
            corig[o] = 0u;
            cbox[o] = make_float4(0.f, 0.f, 0.f, 0.f);
        } else {
            unsigned idx = ~(unsigned)key;               // undo ~idx
            cscore[o] = __uint_as_float(hi);
            corig[o] = idx;
            cbox[o] = boxes[(size_t)b * NN + idx];
        }
    }
}

// ---------------------------------------------------------------- WMMA pairwise suppression
// One wave per (row-tile x 8 col-tile) strip. Column/row boxes staged with
// GLOBAL_LOAD_ASYNC_TO_LDS_B128 (ASYNCcnt), pairwise area-sum a_i + a_j
// = [a_i,1] x [1;a_j]^T via V_WMMA_F32_16X16X4_F32 (A reused across the strip),
// intersection finished in VALU, suppress bits balloted into a KxK/16 bitmatrix.
__global__ __launch_bounds__(32) void rpn_pair_kernel(
    const float4* __restrict__ cbox, unsigned short* __restrict__ sup) {
    int sx = blockIdx.x, rt = blockIdx.y, b = blockIdx.z;
    int ct0 = sx * 8;                         // first column tile of this strip
    if (ct0 + 7 < rt) return;                 // strip fully below diagonal
    __builtin_amdgcn_s_wait_tensorcnt(0);
    int lane = threadIdx.x;

    __shared__ float4 rb[16];                 // row boxes
    __shared__ float4 cb[128];                // 8 col tiles of boxes

    const float4* base = cbox + (size_t)b * KK;

    // async-stage row tile (lanes 0-15, 16B each)
    if (lane < 16) {
        unsigned ldst = (unsigned)(size_t)&rb[lane];
        unsigned long long gsrc = (unsigned long long)(size_t)&base[rt * 16 + lane];
        asm volatile("global_load_async_to_lds_b128 %0, %1, off"
                     :: "v"(ldst), "v"(gsrc) : "memory");
    }
    // async-stage 128 column boxes: 4 x b128 per lane
#pragma unroll
    for (int q = 0; q < 4; ++q) {
        int c = lane + 32 * q;
        unsigned ldst = (unsigned)(size_t)&cb[c];
        unsigned long long gsrc = (unsigned long long)(size_t)&base[ct0 * 16 + c];
        asm volatile("global_load_async_to_lds_b128 %0, %1, off"
                     :: "v"(ldst), "v"(gsrc) : "memory");
    }
    asm volatile("s_wait_asynccnt 0x0" ::: "memory");
    __syncthreads();

    // A (16x4 f32): lanes 0-15 hold M=lane {K0,K1}; lanes 16-31 {K2,K3} = 0.
    // Constant across all 8 tiles of the strip.
    v2f a;
    {
        float4 rB4 = rb[lane & 15];
        if (lane < 16) { a.x = (rB4.z - rB4.x) * (rB4.w - rB4.y); a.y = 1.0f; }
        else           { a.x = 0.0f; a.y = 0.0f; }
    }
    int n = lane & 15;

#pragma unroll
    for (int k = 0; k < 8; ++k) {
        int ct = ct0 + k;
        if (ct < rt) continue;                // uniform (scalar) skip
        float4 cB = cb[k * 16 + n];
        v2f bm;
        if (lane < 16) { bm.x = 1.0f; bm.y = (cB.z - cB.x) * (cB.w - cB.y); }
        else           { bm.x = 0.0f; bm.y = 0.0f; }
        v8f cacc = {};
        v8f d = __builtin_amdgcn_wmma_f32_16x16x4_f32(
            false, a, false, bm, (short)0, cacc, false, false);
#pragma unroll
        for (int r = 0; r < 8; ++r) {
            int m = r + ((lane >> 4) << 3);   // C/D layout: lanes16-31 hold M=r+8
            float4 rB = rb[m];
            float sum = d[r];                 // area_m + area_n from WMMA
            float iy1 = fmaxf(rB.x, cB.x), ix1 = fmaxf(rB.y, cB.y);
            float iy2 = fminf(rB.z, cB.z), ix2 = fminf(rB.w, cB.w);
            float inter = fmaxf(iy2 - iy1, 0.0f) * fmaxf(ix2 - ix1, 0.0f);
            // iou > thr  <=>  inter > thr*(sum - inter + eps)
            bool sbit = inter > IOU_THR * (sum - inter + 1e-8f);
            unsigned mask = (unsigned)__ballot(sbit);   // uniform across wave32
            if (lane == 0) {
                int row0 = rt * 16 + r;
                int row1 = rt * 16 + r + 8;
                sup[((size_t)b * KK + row0) * KW + ct] = (unsigned short)(mask & 0xFFFFu);
                sup[((size_t)b * KK + row1) * KW + ct] = (unsigned short)(mask >> 16);
            }
        }
    }
}

// ---------------------------------------------------------------- serial greedy scan
__global__ __launch_bounds__(256) void rpn_greedy_kernel(
    const float* __restrict__ cscore, const unsigned int* __restrict__ corig,
    const float4* __restrict__ cbox, const unsigned short* __restrict__ sup,
    const float* __restrict__ logits, float* __restrict__ out) {
    int b = blockIdx.x, tid = threadIdx.x;
    __shared__ unsigned int s_alive[KW];
    __shared__ int s_sel;
    if (tid < KW) s_alive[tid] = 0xFFFFu;
    if (tid == 0) s_sel = 0;
    __syncthreads();

    float* o0 = out;                                   // (B,2000,5)
    float* o1 = out + BB * OUTN * 5;                   // (B,2000,2)
    float* o2 = out + BB * OUTN * 5 + BB * OUTN * 2;   // (B,2000,3)

    for (int i = 0; i < KK; ++i) {
        if (s_sel >= OUTN) break;
        int w = i >> 4, bit = i & 15;
        bool alive = (s_alive[w] >> bit) & 1u;
        float sc = cscore[(size_t)b * KK + i];
        bool pick = alive && sc > 0.0f;                // uniform across block
        if (pick) {
            if (tid == 0) {
                int slot = s_sel;
                float4 bx = cbox[(size_t)b * KK + i];
                unsigned orig = corig[(size_t)b * KK + i];
                float l0 = logits[((size_t)b * NN + orig) * 2 + 0];
                float l1 = logits[((size_t)b * NN + orig) * 2 + 1];
                float* q0 = o0 + ((size_t)b * OUTN + slot) * 5;
                q0[0] = bx.x; q0[1] = bx.y; q0[2] = bx.z; q0[3] = bx.w; q0[4] = 1.0f;
                float* q1 = o1 + ((size_t)b * OUTN + slot) * 2;
                q1[0] = sc; q1[1] = 1.0f;
                float* q2 = o2 + ((size_t)b * OUTN + slot) * 3;
                q2[0] = l0; q2[1] = l1; q2[2] = 1.0f;
            }
            const unsigned short* srow = sup + ((size_t)b * KK + i) * KW;
            int w2 = tid;                              // 256 threads == KW words
            if (w2 >= w) {
                unsigned m = srow[w2];
                if (w2 == w) m &= ~((2u << bit) - 1u); // drop j <= i
                s_alive[w2] &= ~m;
            }
            __syncthreads();
            if (tid == 0) s_sel++;
        }
        __syncthreads();
    }
}

// ---------------------------------------------------------------- host
extern "C" void kernel_launch(void* const* d_in, const int* in_sizes, int n_in,
                              void* d_out, int out_size, void* d_ws, size_t ws_size,
                              hipStream_t stream) {
    const float* deltas  = (const float*)d_in[0];
    const float* logits  = (const float*)d_in[1];
    const float* anchors = (const float*)d_in[2];
    float* out = (float*)d_out;

    char* p = (char*)d_ws;
    auto take = [&](size_t bytes) -> char* {
        char* r = p;
        p += ((bytes + 255) & ~(size_t)255);
        return r;
    };
    float4* boxes              = (float4*)take((size_t)BB * NN * 16);
    float* live                = (float*)take((size_t)BB * NN * 4);
    unsigned int* hist         = (unsigned int*)take((size_t)BB * 256 * 4);
    unsigned int* cnt          = (unsigned int*)take((size_t)BB * 4);
    unsigned int* cutoff       = (unsigned int*)take((size_t)BB * 4);
    unsigned long long* ckey   = (unsigned long long*)take((size_t)BB * KK * 8);
    float* cscore              = (float*)take((size_t)BB * KK * 4);
    unsigned int* corig        = (unsigned int*)take((size_t)BB * KK * 4);
    float4* cbox               = (float4*)take((size_t)BB * KK * 16);
    unsigned short* sup        = (unsigned short*)take((size_t)BB * KK * KW * 2);

    rpn_init_kernel<<<(BB * OUTN * 10 + 255) / 256, 256, 0, stream>>>(out, hist, cnt);
    rpn_decode_kernel<<<(BB * NN) / 256, 256, 0, stream>>>(
        (const float4*)deltas, logits, (const float4*)anchors, boxes, live, hist);
    rpn_cutoff_kernel<<<1, 32, 0, stream>>>(hist, cutoff);
    rpn_compact_kernel<<<(BB * NN) / 256, 256, 0, stream>>>(live, cutoff, cnt, ckey);
    rpn_sort_kernel<<<BB, 1024, 0, stream>>>(ckey, cnt, boxes, cscore, corig, cbox);
    dim3 pg(KK / 16 / 8, KK / 16, BB);   // 32 strips x 256 row tiles x B
    rpn_pair_kernel<<<pg, 32, 0, stream>>>(cbox, sup);
    rpn_greedy_kernel<<<BB, 256, 0, stream>>>(cscore, corig, cbox, sup, logits, out);
}